// GCN4Rec_13142599925973
// MI455X (gfx1250) — compile-verified
//
#include <hip/hip_runtime.h>
#include <math.h>

typedef float v2f __attribute__((ext_vector_type(2)));
typedef float v8f __attribute__((ext_vector_type(8)));

#define DIM 64
#define HIDDEN 128

// ---------------------------------------------------------------- utilities
__global__ void fill_kernel(float* __restrict__ p, float v, int n) {
  int i = blockIdx.x * blockDim.x + threadIdx.x;
  if (i < n) p[i] = v;
}

// deg initialized to 1.0 (self-loop); add 1 per incoming edge
__global__ void deg_kernel(const int* __restrict__ dst, float* __restrict__ deg, int E) {
  int e = blockIdx.x * blockDim.x + threadIdx.x;
  if (e < E) atomicAdd(&deg[dst[e]], 1.0f);
}

// deg -> dinv in place (deg >= 1 always due to self loops)
__global__ void dinv_kernel(float* __restrict__ deg, int N) {
  int n = blockIdx.x * blockDim.x + threadIdx.x;
  if (n < N) deg[n] = rsqrtf(fmaxf(deg[n], 1.0f));
}

// per-row clip factor: min(1, 1/(||row||_2 + eps)). One wave per 64-float row.
__global__ void rowscale_kernel(const float* __restrict__ emb, float* __restrict__ scale, int N) {
  int t = blockIdx.x * blockDim.x + threadIdx.x;
  int row = t >> 5, lane = t & 31;
  if (row >= N) return;
  const float* r = emb + (size_t)row * DIM + lane * 2;
  float a = r[0], b = r[1];
  float s = a * a + b * b;
  #pragma unroll
  for (int m = 16; m; m >>= 1) s += __shfl_xor(s, m, 32);
  if (lane == 0) scale[row] = fminf(1.0f, 1.0f / (sqrtf(s) + 1e-7f));
}

// ---------------------------------------------------------------- WMMA GEMM
// C[M x NT*16] = (rowscale .* A[M x K]) @ W[K x NT*16], f32 exact via
// V_WMMA_F32_16X16X4_F32. One wave computes a 16-row strip across all NT tiles.
// ISA f32 fragment layouts (7.12.2):
//   A 16x4 : lanes 0-15 -> M=lane; v0 = K+0, v1 = K+1; lanes 16-31 -> K+2/K+3
//   B 4x16 : lanes 0-15 -> N=lane; v0 = row K+0, v1 = row K+1; lanes 16-31 -> K+2/K+3
//   C 16x16: VGPR r -> (M = r | r+8, N = lane&15)
template <int NT, int K>
__global__ void wmma_gemm_kernel(const float* __restrict__ A,
                                 const float* __restrict__ rowscale,
                                 const float* __restrict__ W,
                                 float* __restrict__ C, int M) {
  const int ldc = NT * 16;
  int wid  = (blockIdx.x * blockDim.x + threadIdx.x) >> 5;
  int lane = threadIdx.x & 31;
  int m0 = wid * 16;
  if (m0 >= M) return;                 // wave-uniform exit (EXEC all-1 for WMMA)
  int half = lane >> 4, l15 = lane & 15;
  int arow = m0 + l15;
  if (arow >= M) arow = M - 1;         // clamp (M=100000 is 16-divisible anyway)
  float asc = rowscale ? rowscale[arow] : 1.0f;
  const float* Ar = A + (size_t)arow * K;

  v8f acc[NT] = {};
  for (int k = 0; k < K; k += 4) {
    int ka = k + 2 * half;
    v2f a;
    a.x = Ar[ka] * asc;
    a.y = Ar[ka + 1] * asc;
    #pragma unroll
    for (int t = 0; t < NT; t++) {
      const float* Wp = W + (size_t)ka * ldc + t * 16 + l15;
      v2f b;
      b.x = Wp[0];
      b.y = Wp[ldc];
      acc[t] = __builtin_amdgcn_wmma_f32_16x16x4_f32(
          false, a, false, b, (short)0, acc[t], false, false);
    }
  }
  #pragma unroll
  for (int r = 0; r < 8; r++) {
    int row = m0 + r + 8 * half;
    if (row < M) {
      #pragma unroll
      for (int t = 0; t < NT; t++)
        C[(size_t)row * ldc + t * 16 + l15] = acc[t][r];
    }
  }
}

// ------------------------------------------------------- edge gather/scatter
// out[dst] += y[src] * dinv[src]*dinv[dst]; virtual edges e>=E are self-loops.
// One wave per edge; lane moves COLS/32 consecutive floats.
template <int COLS>
__global__ void scatter_kernel(const float* __restrict__ y, const int* __restrict__ src,
                               const int* __restrict__ dst, const float* __restrict__ dinv,
                               float* __restrict__ out, int E, int N) {
  int t = blockIdx.x * blockDim.x + threadIdx.x;
  int e = t >> 5, lane = t & 31;
  if (e >= E + N) return;
  int s, d;
  if (e < E) { s = src[e]; d = dst[e]; } else { s = e - E; d = s; }
  float nrm = dinv[s] * dinv[d];
  constexpr int FPL = COLS / 32;
  const float* yp = y + (size_t)s * COLS + lane * FPL;
  float* op = out + (size_t)d * COLS + lane * FPL;
  if constexpr (FPL == 4) {
    float4 v = *(const float4*)yp;
    atomicAdd(op + 0, v.x * nrm);
    atomicAdd(op + 1, v.y * nrm);
    atomicAdd(op + 2, v.z * nrm);
    atomicAdd(op + 3, v.w * nrm);
  } else {
    float2 v = *(const float2*)yp;
    atomicAdd(op + 0, v.x * nrm);
    atomicAdd(op + 1, v.y * nrm);
  }
}

__global__ void relu_bias_kernel(float* __restrict__ h, const float* __restrict__ b, int total) {
  int i = blockIdx.x * blockDim.x + threadIdx.x;
  if (i < total) h[i] = fmaxf(h[i] + b[i & (HIDDEN - 1)], 0.0f);
}

// users = clip(user_emb)[u]; logit = sigmoid(dot(users, out2[i] + b2)). Wave/elem.
__global__ void final_kernel(const float* __restrict__ user_emb, const float* __restrict__ out2,
                             const float* __restrict__ b2, const int* __restrict__ u,
                             const int* __restrict__ iidx, float* __restrict__ logit, int B) {
  int t = blockIdx.x * blockDim.x + threadIdx.x;
  int b = t >> 5, lane = t & 31;
  if (b >= B) return;
  int ur = u[b], ir = iidx[b];
  const float* up = user_emb + (size_t)ur * DIM + lane * 2;
  float u0 = up[0], u1 = up[1];
  float ss = u0 * u0 + u1 * u1;
  #pragma unroll
  for (int m = 16; m; m >>= 1) ss += __shfl_xor(ss, m, 32);
  float sc = fminf(1.0f, 1.0f / (sqrtf(ss) + 1e-7f));
  const float* ip = out2 + (size_t)ir * DIM + lane * 2;
  const float* bp = b2 + lane * 2;
  float dot = u0 * sc * (ip[0] + bp[0]) + u1 * sc * (ip[1] + bp[1]);
  #pragma unroll
  for (int m = 16; m; m >>= 1) dot += __shfl_xor(dot, m, 32);
  if (lane == 0) logit[b] = 1.0f / (1.0f + expf(-dot));
}

// ---------------------------------------------------------------- launcher
static inline int cdiv_l(long a, long b) { return (int)((a + b - 1) / b); }

extern "C" void kernel_launch(void* const* d_in, const int* in_sizes, int n_in,
                              void* d_out, int out_size, void* d_ws, size_t ws_size,
                              hipStream_t stream) {
  const float* entity_emb = (const float*)d_in[0];
  const float* user_emb   = (const float*)d_in[1];
  const float* W1 = (const float*)d_in[2];
  const float* b1 = (const float*)d_in[3];
  const float* W2 = (const float*)d_in[4];
  const float* b2 = (const float*)d_in[5];
  const int* u    = (const int*)d_in[6];
  const int* iidx = (const int*)d_in[7];
  const int* edge = (const int*)d_in[8];

  int N = in_sizes[0] / DIM;     // 100000 entities
  int B = in_sizes[6];           // 8192
  int E = in_sizes[8] / 2;       // 1.6M edges
  const int* esrc = edge;
  const int* edst = edge + E;

  // workspace layout (floats): deg/dinv[N] | rowscale[N] | bufA[N*128] | bufB[N*128]
  // bufA: y1 (N*128), later reused as y2 (first N*64) + out2 (second N*64)
  // bufB: layer-1 accumulator, then h in place
  float* ws   = (float*)d_ws;
  float* deg  = ws;
  float* rsc  = ws + (size_t)N;
  float* bufA = ws + 2 * (size_t)N;
  float* bufB = bufA + (size_t)N * HIDDEN;
  float* y1   = bufA;
  float* h    = bufB;
  float* y2   = bufA;
  float* out2 = bufA + (size_t)N * DIM;

  // --- norm precompute (re-done every call: deterministic for graph replay)
  fill_kernel<<<cdiv_l(N, 256), 256, 0, stream>>>(deg, 1.0f, N);
  fill_kernel<<<cdiv_l((long)N * HIDDEN, 256), 256, 0, stream>>>(bufB, 0.0f, N * HIDDEN);
  rowscale_kernel<<<cdiv_l((long)N * 32, 256), 256, 0, stream>>>(entity_emb, rsc, N);
  deg_kernel<<<cdiv_l(E, 256), 256, 0, stream>>>(edst, deg, E);
  dinv_kernel<<<cdiv_l(N, 256), 256, 0, stream>>>(deg, N);

  int gemm_waves = cdiv_l(N, 16);
  int gemm_blocks = cdiv_l((long)gemm_waves * 32, 256);
  int edge_blocks = cdiv_l((long)(E + N) * 32, 256);

  // --- layer 1: y1 = clip(X) @ W1 ; acc = A_norm @ y1 ; h = relu(acc + b1)
  wmma_gemm_kernel<8, 64><<<gemm_blocks, 256, 0, stream>>>(entity_emb, rsc, W1, y1, N);
  scatter_kernel<128><<<edge_blocks, 256, 0, stream>>>(y1, esrc, edst, deg, h, E, N);
  relu_bias_kernel<<<cdiv_l((long)N * HIDDEN, 256), 256, 0, stream>>>(h, b1, N * HIDDEN);

  // --- layer 2: y2 = h @ W2 ; out2 = A_norm @ y2  (b2 folded into final dot)
  wmma_gemm_kernel<4, 128><<<gemm_blocks, 256, 0, stream>>>(h, nullptr, W2, y2, N);
  fill_kernel<<<cdiv_l((long)N * DIM, 256), 256, 0, stream>>>(out2, 0.0f, N * DIM);
  scatter_kernel<64><<<edge_blocks, 256, 0, stream>>>(y2, esrc, edst, deg, out2, E, N);

  // --- batched scoring
  final_kernel<<<cdiv_l((long)B * 32, 256), 256, 0, stream>>>(
      user_emb, out2, b2, u, iidx, (float*)d_out, B);
}